// VoxelMix_13486197310125
// MI455X (gfx1250) — compile-verified
//
#include <hip/hip_runtime.h>
#include <stdint.h>

typedef int v4i __attribute__((ext_vector_type(4)));

#define N_PTS 2000000
#define F_DIM 16
#define B_DIM 8
#define R_DIM 480
#define A_DIM 360
#define Z_DIM 32
#define RKMIN 50
#define CUTN  8

// ---- CDNA5 async global<->LDS helpers (ASYNCcnt path), non-temporal ----
__device__ __forceinline__ void async_load_b128_nt(unsigned lds_off, uint64_t gaddr) {
    asm volatile("global_load_async_to_lds_b128 %0, %1, off th:TH_LOAD_NT"
                 :: "v"(lds_off), "v"(gaddr) : "memory");
}
__device__ __forceinline__ void async_store_b128_nt(uint64_t gaddr, unsigned lds_off) {
    asm volatile("global_store_async_from_lds_b128 %0, %1, off th:TH_STORE_NT"
                 :: "v"(gaddr), "v"(lds_off) : "memory");
}
__device__ __forceinline__ void wait_async0() {
#if __has_builtin(__builtin_amdgcn_s_wait_asynccnt)
    __builtin_amdgcn_s_wait_asynccnt(0);
#else
    asm volatile("s_wait_asynccnt 0x0" ::: "memory");
#endif
}

// ---- Kernel A: 128MB passthrough copy of point_feature via async LDS DMA ----
// 8,000,000 16B chunks = 15,625 blocks x 256 threads x 2 chunks -> no tail,
// no guards, no exec-mask churn. Data never touches VGPRs.
__global__ __launch_bounds__(256) void pf_copy_async(const float* __restrict__ in,
                                                     float* __restrict__ out) {
    __shared__ alignas(16) unsigned char smem[256 * 2 * 16];
    const unsigned tid   = threadIdx.x;
    const unsigned lbase = (unsigned)(size_t)(&smem[0]);   // low 32 bits = LDS offset
    const uint64_t gin   = (uint64_t)in;
    const uint64_t gout  = (uint64_t)out;
    const unsigned base  = blockIdx.x * 512u;

#pragma unroll
    for (int c = 0; c < 2; ++c) {
        unsigned idx = base + (unsigned)c * 256u + tid;
        async_load_b128_nt(lbase + ((unsigned)c * 256u + tid) * 16u,
                           gin + (uint64_t)idx * 16u);
    }
    wait_async0();   // loads must land in LDS before dependent stores read it
#pragma unroll
    for (int c = 0; c < 2; ++c) {
        unsigned idx = base + (unsigned)c * 256u + tid;
        async_store_b128_nt(gout + (uint64_t)idx * 16u,
                            lbase + ((unsigned)c * 256u + tid) * 16u);
    }
    // S_ENDPGM performs an implicit wait-idle -> async stores complete.
}

// ---- Kernel B: rewrite point_coord column 0 via inverse permutation ----
// bounds[c] = floor(45c/2) => bucket(th) = min((2*th+1)/45, 7)
__global__ __launch_bounds__(256) void pc_mix(const int* __restrict__ pc,
                                              const int* __restrict__ urand,
                                              int* __restrict__ out) {
    __shared__ int s_inv[CUTN * B_DIM];
    const unsigned tid = threadIdx.x;
    if (tid < CUTN * B_DIM) {
        int c = (int)(tid >> 3);
        int j = (int)(tid & 7u);
        int val = urand[tid];              // u_rand_index[c][j]
        s_inv[c * B_DIM + val] = j;        // inverse permutation
    }
    __syncthreads();

    int i = (int)(blockIdx.x * 256u + tid);
    if (i >= N_PTS) return;

    int b0 = pc[3 * i + 0];
    int r  = pc[3 * i + 1];
    int th = pc[3 * i + 2];

    unsigned area = (unsigned)(2 * th + 1) / 45u;
    if (area > 7u) area = 7u;
    bool valid = (th >= 0) && (th < 180) && (r >= RKMIN);
    int nb = valid ? s_inv[area * B_DIM + (unsigned)b0] : b0;

    out[3 * i + 0] = nb;
    out[3 * i + 1] = r;
    out[3 * i + 2] = th;
}

// ---- Kernel C: batch-axis row gather of voxel_label (128B rows, b128/lane, NT) ----
__global__ __launch_bounds__(256) void vl_mix(const int* __restrict__ vl,
                                              const int* __restrict__ urand,
                                              int* __restrict__ out) {
    __shared__ int s_u[CUTN * B_DIM];
    const unsigned tid = threadIdx.x;
    if (tid < CUTN * B_DIM) s_u[tid] = urand[tid];
    __syncthreads();

    const unsigned gid  = blockIdx.x * 256u + tid;
    const unsigned row  = gid >> 3;       // (b, r, a) row index
    const unsigned part = gid & 7u;       // 8 lanes x 16B = 128B row
    if (row >= (unsigned)(B_DIM * R_DIM * A_DIM)) return;

    unsigned a = row % A_DIM;
    unsigned t = row / A_DIM;
    unsigned r = t % R_DIM;
    unsigned b = t / R_DIM;

    unsigned srcb = b;
    if (r >= RKMIN && a < 180u) {
        unsigned area = (2u * a + 1u) / 45u;
        if (area > 7u) area = 7u;
        srcb = (unsigned)s_u[area * B_DIM + b];
    }

    size_t srcoff = ((size_t)((srcb * R_DIM + r) * A_DIM + a)) * Z_DIM + (size_t)part * 4u;
    size_t dstoff = (size_t)row * Z_DIM + (size_t)part * 4u;

    v4i v = __builtin_nontemporal_load((const v4i*)(vl + srcoff));
    __builtin_nontemporal_store(v, (v4i*)(out + dstoff));
}

extern "C" void kernel_launch(void* const* d_in, const int* in_sizes, int n_in,
                              void* d_out, int out_size, void* d_ws, size_t ws_size,
                              hipStream_t stream) {
    const float* pf = (const float*)d_in[0];
    const int*   pc = (const int*)d_in[1];
    const int*   vl = (const int*)d_in[2];
    const int*   ur = (const int*)d_in[3];

    float* out_pf = (float*)d_out;
    int*   out_pc = (int*)((char*)d_out + (size_t)N_PTS * F_DIM * sizeof(float));
    int*   out_vl = out_pc + (size_t)N_PTS * 3;

    const int nchunk = N_PTS * F_DIM / 4;                 // 8,000,000 x 16B
    pf_copy_async<<<nchunk / 512, 256, 0, stream>>>(pf, out_pf);   // 15,625 blocks, exact

    pc_mix<<<(N_PTS + 255) / 256, 256, 0, stream>>>(pc, ur, out_pc);

    const unsigned rows = B_DIM * R_DIM * A_DIM;          // 1,382,400
    vl_mix<<<(rows * 8 + 255) / 256, 256, 0, stream>>>(vl, ur, out_vl);
}